// CreateOverlappingWindows_25666724560969
// MI455X (gfx1250) — compile-verified
//
#include <hip/hip_runtime.h>
#include <hip/hip_bf16.h>
#include <stdint.h>

// out[b, t, w*26 + c] = xpad[b, t + w, c],  pad = 9 frames each side.
// Output row t == contiguous input span starting at element (t-9)*26.
// Pure data movement: store-bandwidth bound (~253 MB stores @ 23.3 TB/s ~ 11us).

#define N_CONTEXT 9
#define N_INPUT   26
#define WINDOW    (2 * N_CONTEXT + 1)   // 19
#define OUTC      (WINDOW * N_INPUT)    // 494
#define TDIM      2000
#define BDIM      64
#define TROWS     64                    // output rows per block
#define LDSROWS   (TROWS + 2 * N_CONTEXT)   // 82
#define LDSN      (LDSROWS * N_INPUT)       // 2132 floats = 8528 B

typedef float v2f __attribute__((ext_vector_type(2)));   // native vector: OK for NT builtin

__global__ __launch_bounds__(256)
void CreateOverlappingWindows_25666724560969_kernel(const float* __restrict__ x,
                                                    float* __restrict__ out) {
    __shared__ float lds[LDSN];

    const int b   = blockIdx.y;
    const int t0  = blockIdx.x * TROWS;
    const int tid = threadIdx.x;

    // ---- Phase 1: zero LDS (provides the SAME-padding zeros at batch edges)
    for (int j = tid; j < LDSN; j += 256) lds[j] = 0.0f;
    __syncthreads();  // DS stores complete before async engine writes LDS

    // ---- Phase 2: async global->LDS copy of the valid input span
    // LDS element j  <->  global element gstart + j
    const long long batch_base = (long long)b * (TDIM * N_INPUT);
    const long long gstart     = batch_base + (long long)(t0 - N_CONTEXT) * N_INPUT;
    const long long bend       = batch_base + (long long)TDIM * N_INPUT;
    long long s0 = gstart < batch_base ? batch_base : gstart;
    long long s1 = (gstart + LDSN) < bend ? (gstart + LDSN) : bend;
    const int n  = (int)(s1 - s0);       // always even
    const int l0 = (int)(s0 - gstart);   // always even
    const unsigned lds_base = (unsigned)(uintptr_t)(&lds[0]);  // low 32 bits = LDS offset
    const float* gp = x + s0;
    const int npairs = n >> 1;           // B64 transfers (8B aligned both sides)
    for (int j = tid; j < npairs; j += 256) {
        const unsigned laddr = lds_base + (unsigned)((l0 + 2 * j) * 4);
        const float*   ga    = gp + 2 * j;
        asm volatile("global_load_async_to_lds_b64 %0, %1, off"
                     :: "v"(laddr), "v"(ga) : "memory");
    }
    asm volatile("s_wait_asynccnt 0x0" ::: "memory");
    __syncthreads();

    // ---- Phase 3: stream output rows; row t = lds[i*26 .. i*26+493]
    // Non-temporal B64 stores: don't pollute L2 with the 253 MB write stream.
    const int wave = tid >> 5;
    const int lane = tid & 31;
    for (int i = wave; i < TROWS; i += 8) {
        const int t = t0 + i;
        if (t >= TDIM) break;                       // uniform per wave
        const v2f* src = (const v2f*)(lds + i * N_INPUT);
        v2f*       dst = (v2f*)(out + (size_t)(b * TDIM + t) * OUTC);
        #pragma unroll
        for (int k2 = lane; k2 < OUTC / 2; k2 += 32) {
            v2f v = src[k2];
            __builtin_nontemporal_store(v, dst + k2);
        }
    }
}

extern "C" void kernel_launch(void* const* d_in, const int* in_sizes, int n_in,
                              void* d_out, int out_size, void* d_ws, size_t ws_size,
                              hipStream_t stream) {
    const float* x  = (const float*)d_in[0];
    float*      out = (float*)d_out;
    dim3 grid((TDIM + TROWS - 1) / TROWS, BDIM);   // 32 x 64 = 2048 blocks
    CreateOverlappingWindows_25666724560969_kernel<<<grid, 256, 0, stream>>>(x, out);
}